// IzhikevichNeuron_90314572300837
// MI455X (gfx1250) — compile-verified
//
#include <hip/hip_runtime.h>
#include <stdint.h>

// Izhikevich scan: I_seq [B=16, T=1000, D=4096] fp32 -> spikes same shape.
// Memory-bound (512 MB traffic, ~22us roofline at 23.3 TB/s). Strategy:
// one thread per neuron, deep async global->LDS prefetch FIFO (DEPTH=32
// outstanding per wave, ~8 MB in flight chip-wide) to cover HBM latency,
// 4-step unrolled consumption (1 s_wait_asynccnt per 4 steps),
// non-temporal spike stores so output doesn't evict streamed input from L2.

#define BLOCK 256
#define DEPTH 32           // outstanding async copies per wave (ASYNCcnt max 63)
#define UNROLL 4           // steps consumed per s_wait_asynccnt
#define T_STEPS 1000
#define D_DIM 4096
#define B_DIM 16

typedef __attribute__((address_space(1))) int ai1_t;   // global (AS1) int
typedef __attribute__((address_space(3))) int ai3_t;   // LDS (AS3) int
typedef __attribute__((address_space(3))) float af3_t;

__device__ __forceinline__ void async_copy_b32(const float* g, float* l) {
#if defined(__HIP_DEVICE_COMPILE__)
#if __has_builtin(__builtin_amdgcn_global_load_async_to_lds_b32)
    // (global src AS1, LDS dst AS3, imm offset, imm cpol)
    __builtin_amdgcn_global_load_async_to_lds_b32((ai1_t*)g, (ai3_t*)l, 0, 0);
#else
    uint32_t laddr = (uint32_t)(uintptr_t)(af3_t*)l;
    asm volatile("global_load_async_to_lds_b32 %0, %1, off"
                 :: "v"(laddr), "v"(g) : "memory");
#endif
#endif
}

template <int N>
__device__ __forceinline__ void wait_asynccnt() {
#if defined(__HIP_DEVICE_COMPILE__)
#if __has_builtin(__builtin_amdgcn_s_wait_asynccnt)
    __builtin_amdgcn_s_wait_asynccnt(N);
#else
    asm volatile("s_wait_asynccnt %0" :: "n"(N) : "memory");
#endif
#endif
}

// One Izhikevich step. Updates v,u in place, returns spike (0.0/1.0).
__device__ __forceinline__ float izh_step(float& v, float& u, float it) {
    // dv = 0.04 v^2 + 5 v + 140 - u + I
    float dv = fmaf(0.04f * v, v, fmaf(5.0f, v, 140.0f)) - u + it;
    v = fmaf(0.2f, dv, v);                       // v += DT*dv
    float du = 0.02f * fmaf(0.2f, v, -u);        // du = A*(B*v_new - u)
    u = fmaf(0.2f, du, u);                       // u += DT*du
    float spk = (v >= 30.0f) ? 1.0f : 0.0f;
    if (spk > 0.0f) { v = -65.0f; u = u + 6.0f; }
    return spk;
}

__global__ __launch_bounds__(BLOCK) void izh_kernel(const float* __restrict__ I,
                                                    float* __restrict__ out) {
    __shared__ float buf[DEPTH * BLOCK];

    const int tid = threadIdx.x;
    const int n   = blockIdx.x * BLOCK + tid;    // neuron id in [0, B*D)
    const int b   = n >> 12;                     // n / 4096
    const int d   = n & (D_DIM - 1);             // n % 4096
    const size_t base = (size_t)b * T_STEPS * D_DIM + d;
    const float* gin  = I + base;
    float*       gout = out + base;

    // Prologue: queue DEPTH async copies (each instr moves 128B per wave).
    #pragma unroll
    for (int k = 0; k < DEPTH; ++k) {
        async_copy_b32(gin + (size_t)k * D_DIM, &buf[k * BLOCK + tid]);
    }

    float v = -65.0f;
    float u = 0.2f * -65.0f;

    const int t_main = T_STEPS - DEPTH;          // 968, divisible by UNROLL
    for (int t = 0; t < t_main; t += UNROLL) {
        // In-order completion: asynccnt <= DEPTH-UNROLL => slots t..t+3 done.
        wait_asynccnt<DEPTH - UNROLL>();
        #pragma unroll
        for (int j = 0; j < UNROLL; ++j) {
            const int tj   = t + j;
            const int slot = tj & (DEPTH - 1);
            float it  = buf[slot * BLOCK + tid]; // ds_load_b32, own lane only
            float spk = izh_step(v, u, it);
            __builtin_nontemporal_store(spk, gout + (size_t)tj * D_DIM);
            // Refill this slot with step tj+DEPTH (value already consumed,
            // so the ds_load retired before the async write can land).
            async_copy_b32(gin + (size_t)(tj + DEPTH) * D_DIM,
                           &buf[slot * BLOCK + tid]);
        }
    }

    // Drain: everything outstanding is ours; one wait, then consume freely.
    wait_asynccnt<0>();
    #pragma unroll 4
    for (int t = t_main; t < T_STEPS; ++t) {
        const int slot = t & (DEPTH - 1);
        float it  = buf[slot * BLOCK + tid];
        float spk = izh_step(v, u, it);
        __builtin_nontemporal_store(spk, gout + (size_t)t * D_DIM);
    }
}

extern "C" void kernel_launch(void* const* d_in, const int* in_sizes, int n_in,
                              void* d_out, int out_size, void* d_ws, size_t ws_size,
                              hipStream_t stream) {
    (void)in_sizes; (void)n_in; (void)d_ws; (void)ws_size; (void)out_size;
    const float* I = (const float*)d_in[0];
    float* out = (float*)d_out;
    const int n_neurons = B_DIM * D_DIM;         // 65536
    dim3 grid(n_neurons / BLOCK);                // 256 blocks
    dim3 block(BLOCK);
    izh_kernel<<<grid, block, 0, stream>>>(I, out);
}